// dMaSIF_87136296501945
// MI455X (gfx1250) — compile-verified
//
#include <hip/hip_runtime.h>
#include <hip/hip_bf16.h>

typedef __attribute__((ext_vector_type(2))) float v2f;
typedef __attribute__((ext_vector_type(8))) float v8f;

__device__ __forceinline__ float lrelu(float x) { return x >= 0.0f ? x : 0.2f * x; }

// ---------------------------------------------------------------------------
// Kernel 1: atomtypes 6 -> 6 -> 6 MLP (tiny; per-thread)
// ---------------------------------------------------------------------------
__global__ __launch_bounds__(256) void tt_mlp_kernel(
    const float* __restrict__ at, const float* __restrict__ w1,
    const float* __restrict__ b1, const float* __restrict__ w2,
    const float* __restrict__ b2, float* __restrict__ out, int n)
{
    int i = blockIdx.x * blockDim.x + threadIdx.x;
    if (i >= n) return;
    float x[6], h[6];
#pragma unroll
    for (int c = 0; c < 6; ++c) x[c] = at[i * 6 + c];
#pragma unroll
    for (int o = 0; o < 6; ++o) {
        float acc = b1[o];
#pragma unroll
        for (int c = 0; c < 6; ++c) acc += x[c] * w1[o * 6 + c];
        h[o] = lrelu(acc);
    }
#pragma unroll
    for (int o = 0; o < 6; ++o) {
        float acc = b2[o];
#pragma unroll
        for (int c = 0; c < 6; ++c) acc += h[c] * w2[o * 6 + c];
        out[i * 6 + o] = acc;
    }
}

// ---------------------------------------------------------------------------
// Kernel 2: kNN via WMMA f32 16x16x4 (full-fp32).
// A fragments are pre-scaled by -2, and C is seeded with |q|^2 + |t|^2, so
// each v_wmma_f32_16x16x4_f32 directly emits a 16x16 tile of squared
// distances: D = (-2A) x B + (qn + tn). One wave owns 32 queries (2 C tiles),
// stages d2 in padded LDS (stride 17, conflict-free row reads), and each lane
// runs a fully-unrolled register top-K insertion for its own query row.
// KTOT = 17 for atom-atom (slot 0 == self, dropped), 16 for point-atom.
// nt must be a multiple of 16 (8000 is).
// ---------------------------------------------------------------------------
template <int KTOT, bool DROP_SELF>
__global__ __launch_bounds__(256) void knn_wmma_kernel(
    const float* __restrict__ Q, int nq,
    const float* __restrict__ T, int nt,
    int* __restrict__ oidx, float* __restrict__ odist)
{
    __shared__ float s_tile[8][32][17];   // per-wave 32x16 d2 stage, pad->17

    const int lane  = threadIdx.x & 31;
    const int wave  = threadIdx.x >> 5;
    const int qbase = (blockIdx.x * 8 + wave) * 32;
    if (qbase >= nq) return;              // uniform per wave

    // ---- my query (clamped; nq is a multiple of 32 here anyway) ----
    const int qi = qbase + lane;
    const int qc = (qi < nq) ? qi : (nq - 1);
    const float qx = Q[qc * 3 + 0];
    const float qy = Q[qc * 3 + 1];
    const float qz = Q[qc * 3 + 2];
    const float qn = qx * qx + qy * qy + qz * qz;

    // ---- build the two A fragments (rows 0..15 and 16..31) via shuffles,
    //      pre-scaled by -2 so the WMMA computes -2 q.t directly ----
    // A layout (16x4 f32): v0: lanes0-15 K=0, lanes16-31 K=2
    //                      v1: lanes0-15 K=1, lanes16-31 K=3 (pad 0)
    const int lsrc = lane & 15;
    const float zlo = __shfl(qz, lsrc);        // z of query row lane%16
    const float xhi = __shfl(qx, lsrc + 16);   // coords of query row lane%16+16
    const float yhi = __shfl(qy, lsrc + 16);
    v2f a0, a1;
    a0.x = -2.0f * ((lane < 16) ? qx : zlo);
    a0.y = -2.0f * ((lane < 16) ? qy : 0.0f);
    a1.x = -2.0f * ((lane < 16) ? xhi : qz);
    a1.y = -2.0f * ((lane < 16) ? yhi : 0.0f);

    // ---- |q|^2 per C-tile row, hoisted out of the loop (fixed per lane) ----
    const int half = (lane >> 4) << 3;         // 0 or 8
    float qnA[8], qnB[8];
#pragma unroll
    for (int r = 0; r < 8; ++r) {
        qnA[r] = __shfl(qn, half + r);
        qnB[r] = __shfl(qn, 16 + half + r);
    }

    // ---- top-K state in registers (sorted ascending) ----
    float bd[KTOT];
    int   bi[KTOT];
#pragma unroll
    for (int t = 0; t < KTOT; ++t) { bd[t] = 3.0e38f; bi[t] = 0; }

    float* const wtile = &s_tile[wave][0][0];
    const int colh = lane & 15;

    for (int tb = 0; tb < nt; tb += 16) {
        // prefetch the tile 8 iterations ahead
        int pf = tb + 128; pf = (pf < nt) ? pf : (nt - 16);
        __builtin_prefetch(&T[(size_t)pf * 3], 0, 1);

        // ---- B fragment: targets tb..tb+15, both lane halves load same col
        const int tcol = tb + colh;
        const float tx = T[tcol * 3 + 0];
        const float ty = T[tcol * 3 + 1];
        const float tz = T[tcol * 3 + 2];
        const float tn = tx * tx + ty * ty + tz * tz;
        v2f bf;
        bf.x = (lane < 16) ? tx : tz;
        bf.y = (lane < 16) ? ty : 0.0f;

        // ---- seed accumulators with |q|^2 + |t|^2 ----
        v8f cc0, cc1;
#pragma unroll
        for (int r = 0; r < 8; ++r) { cc0[r] = qnA[r] + tn; cc1[r] = qnB[r] + tn; }

        // ---- d2 tiles straight out of the matrix engine ----
        v8f c0 = __builtin_amdgcn_wmma_f32_16x16x4_f32(
            false, a0, false, bf, (short)0, cc0, false, false);
        v8f c1 = __builtin_amdgcn_wmma_f32_16x16x4_f32(
            false, a1, false, bf, (short)0, cc1, false, false);

        // ---- stage to LDS (row-major, pad 17) ----
#pragma unroll
        for (int r = 0; r < 8; ++r) {
            wtile[(half + r) * 17 + colh]      = c0[r];
            wtile[(16 + half + r) * 17 + colh] = c1[r];
        }
        // wave-synchronous: same-wave LDS ops are in order, no barrier needed.

        // ---- lane l scans its own query row l and inserts into top-K ----
        const float* wrow = &s_tile[wave][lane][0];
#pragma unroll
        for (int j = 0; j < 16; ++j) {
            const float d = wrow[j];
            if (d < bd[KTOT - 1]) {
                bd[KTOT - 1] = d;
                bi[KTOT - 1] = tb + j;
#pragma unroll
                for (int t = KTOT - 1; t > 0; --t) {
                    if (bd[t] < bd[t - 1]) {
                        float td = bd[t]; bd[t] = bd[t - 1]; bd[t - 1] = td;
                        int   ti = bi[t]; bi[t] = bi[t - 1]; bi[t - 1] = ti;
                    }
                }
            }
        }
    }

    // ---- emit 16 neighbors; recompute exact squared distance (ref does) ----
    if (qi < nq) {
#pragma unroll
        for (int j = 0; j < 16; ++j) {
            const int s  = DROP_SELF ? (j + 1) : j;
            const int id = bi[s];
            const float dx = qx - T[id * 3 + 0];
            const float dy = qy - T[id * 3 + 1];
            const float dz = qz - T[id * 3 + 2];
            oidx[(size_t)qi * 16 + j]  = id;
            odist[(size_t)qi * 16 + j] = dx * dx + dy * dy + dz * dz;
        }
    }
}

// ---------------------------------------------------------------------------
// Kernel 3: message-passing block.
// feat = [self(6), nb(6), dist(1)] (H=13) -> W1(13x13)+lrelu -> W2(6x13),
// sum over 16 neighbors, group_norm(groups=2), lrelu, residual.
// Weights staged in LDS; per-thread compute (dims too small for WMMA tiles).
// ---------------------------------------------------------------------------
__global__ __launch_bounds__(256) void mp_block_kernel(
    const float* __restrict__ self_feat,   // n x 6
    const float* __restrict__ nb_feat,     // nsrc x 6
    const int*   __restrict__ idx,         // n x 16
    const float* __restrict__ dists,       // n x 16
    const float* __restrict__ w1, const float* __restrict__ b1,   // 13x13, 13
    const float* __restrict__ w2, const float* __restrict__ b2,   // 6x13, 6
    const float* __restrict__ gamma, const float* __restrict__ beta,
    float* __restrict__ out, int n)
{
    __shared__ float sW1[169], sW2[78], sB1[13], sB2[6], sG[6], sBt[6];
    for (int t = threadIdx.x; t < 169; t += blockDim.x) sW1[t] = w1[t];
    for (int t = threadIdx.x; t < 78;  t += blockDim.x) sW2[t] = w2[t];
    for (int t = threadIdx.x; t < 13;  t += blockDim.x) sB1[t] = b1[t];
    for (int t = threadIdx.x; t < 6;   t += blockDim.x) {
        sB2[t] = b2[t]; sG[t] = gamma[t]; sBt[t] = beta[t];
    }
    __syncthreads();

    const int i = blockIdx.x * blockDim.x + threadIdx.x;
    if (i >= n) return;

    float s[6];
#pragma unroll
    for (int c = 0; c < 6; ++c) s[c] = self_feat[(size_t)i * 6 + c];

    float msg[6] = {0.f, 0.f, 0.f, 0.f, 0.f, 0.f};
    for (int k = 0; k < 16; ++k) {
        const int j = idx[(size_t)i * 16 + k];
        float f[13];
#pragma unroll
        for (int c = 0; c < 6; ++c) f[c] = s[c];
#pragma unroll
        for (int c = 0; c < 6; ++c) f[6 + c] = nb_feat[(size_t)j * 6 + c];
        f[12] = dists[(size_t)i * 16 + k];

        float h[13];
#pragma unroll
        for (int o = 0; o < 13; ++o) {
            float acc = sB1[o];
#pragma unroll
            for (int c = 0; c < 13; ++c) acc += f[c] * sW1[o * 13 + c];
            h[o] = lrelu(acc);
        }
#pragma unroll
        for (int o = 0; o < 6; ++o) {
            float acc = sB2[o];
#pragma unroll
            for (int c = 0; c < 13; ++c) acc += h[c] * sW2[o * 13 + c];
            msg[o] += acc;
        }
    }

    // group norm: 2 groups of 3 channels
#pragma unroll
    for (int g = 0; g < 2; ++g) {
        const float m0 = msg[3 * g], m1 = msg[3 * g + 1], m2 = msg[3 * g + 2];
        const float mu = (m0 + m1 + m2) * (1.0f / 3.0f);
        const float d0 = m0 - mu, d1 = m1 - mu, d2 = m2 - mu;
        const float var = (d0 * d0 + d1 * d1 + d2 * d2) * (1.0f / 3.0f);
        const float rs = rsqrtf(var + 1e-5f);
#pragma unroll
        for (int c = 0; c < 3; ++c) {
            const int ch = 3 * g + c;
            const float xn = (msg[ch] - mu) * rs;
            out[(size_t)i * 6 + ch] = s[ch] + lrelu(xn * sG[ch] + sBt[ch]);
        }
    }
}

// ---------------------------------------------------------------------------
// Kernel 4: fill buffer with 1.0f (emb init)
// ---------------------------------------------------------------------------
__global__ __launch_bounds__(256) void fill_ones_kernel(float* __restrict__ p, int n)
{
    int i = blockIdx.x * blockDim.x + threadIdx.x;
    if (i < n) p[i] = 1.0f;
}

// ---------------------------------------------------------------------------
extern "C" void kernel_launch(void* const* d_in, const int* in_sizes, int n_in,
                              void* d_out, int out_size, void* d_ws, size_t ws_size,
                              hipStream_t stream)
{
    const float* xyz      = (const float*)d_in[0];
    const float* atom_xyz = (const float*)d_in[1];
    const float* atomtyp  = (const float*)d_in[2];
    const float* tt_w1 = (const float*)d_in[5];
    const float* tt_b1 = (const float*)d_in[6];
    const float* tt_w2 = (const float*)d_in[7];
    const float* tt_b2 = (const float*)d_in[8];
    const float* aa_w1 = (const float*)d_in[9];
    const float* aa_b1 = (const float*)d_in[10];
    const float* aa_w2 = (const float*)d_in[11];
    const float* aa_b2 = (const float*)d_in[12];
    const float* aa_g  = (const float*)d_in[13];
    const float* aa_bt = (const float*)d_in[14];
    const float* em_w1 = (const float*)d_in[15];
    const float* em_b1 = (const float*)d_in[16];
    const float* em_w2 = (const float*)d_in[17];
    const float* em_b2 = (const float*)d_in[18];
    const float* em_g  = (const float*)d_in[19];
    const float* em_bt = (const float*)d_in[20];

    const int n_pts = in_sizes[0] / 3;   // 20000
    const int n_at  = in_sizes[1] / 3;   // 8000

    // ---- workspace carving (256B aligned regions) ----
    size_t off = 0;
    auto carve = [&](size_t bytes) -> void* {
        void* p = (char*)d_ws + off;
        off += (bytes + 255) & ~(size_t)255;
        return p;
    };
    float* at0   = (float*)carve((size_t)n_at  * 6  * 4);
    float* at1   = (float*)carve((size_t)n_at  * 6  * 4);
    int*   idxA  = (int*)  carve((size_t)n_at  * 16 * 4);
    float* dstA  = (float*)carve((size_t)n_at  * 16 * 4);
    int*   idxP  = (int*)  carve((size_t)n_pts * 16 * 4);
    float* dstP  = (float*)carve((size_t)n_pts * 16 * 4);
    float* emb0  = (float*)carve((size_t)n_pts * 6  * 4);
    float* emb1  = (float*)carve((size_t)n_pts * 6  * 4);
    (void)ws_size; (void)n_in; (void)out_size;

    const int B = 256;
    const int gAt  = (n_at  + B - 1) / B;
    const int gPts = (n_pts + B - 1) / B;

    // 1) atom type MLP
    tt_mlp_kernel<<<gAt, B, 0, stream>>>(atomtyp, tt_w1, tt_b1, tt_w2, tt_b2, at0, n_at);

    // 2) atom-atom kNN (top-17, drop self)
    knn_wmma_kernel<17, true><<<gAt, B, 0, stream>>>(atom_xyz, n_at, atom_xyz, n_at, idxA, dstA);

    // 3) three atom MP blocks (ping-pong at0 <-> at1)
    mp_block_kernel<<<gAt, B, 0, stream>>>(at0, at0, idxA, dstA,
        aa_w1 + 0 * 169, aa_b1 + 0 * 13, aa_w2 + 0 * 78, aa_b2 + 0 * 6,
        aa_g + 0 * 6, aa_bt + 0 * 6, at1, n_at);
    mp_block_kernel<<<gAt, B, 0, stream>>>(at1, at1, idxA, dstA,
        aa_w1 + 1 * 169, aa_b1 + 1 * 13, aa_w2 + 1 * 78, aa_b2 + 1 * 6,
        aa_g + 1 * 6, aa_bt + 1 * 6, at0, n_at);
    mp_block_kernel<<<gAt, B, 0, stream>>>(at0, at0, idxA, dstA,
        aa_w1 + 2 * 169, aa_b1 + 2 * 13, aa_w2 + 2 * 78, aa_b2 + 2 * 6,
        aa_g + 2 * 6, aa_bt + 2 * 6, at1, n_at);
    // final atom features in at1

    // 4) point-atom kNN (top-16)
    knn_wmma_kernel<16, false><<<gPts, B, 0, stream>>>(xyz, n_pts, atom_xyz, n_at, idxP, dstP);

    // 5) emb = ones; three point MP blocks (nb source = at1)
    fill_ones_kernel<<<(n_pts * 6 + B - 1) / B, B, 0, stream>>>(emb0, n_pts * 6);
    mp_block_kernel<<<gPts, B, 0, stream>>>(emb0, at1, idxP, dstP,
        em_w1 + 0 * 169, em_b1 + 0 * 13, em_w2 + 0 * 78, em_b2 + 0 * 6,
        em_g + 0 * 6, em_bt + 0 * 6, emb1, n_pts);
    mp_block_kernel<<<gPts, B, 0, stream>>>(emb1, at1, idxP, dstP,
        em_w1 + 1 * 169, em_b1 + 1 * 13, em_w2 + 1 * 78, em_b2 + 1 * 6,
        em_g + 1 * 6, em_bt + 1 * 6, emb0, n_pts);
    mp_block_kernel<<<gPts, B, 0, stream>>>(emb0, at1, idxP, dstP,
        em_w1 + 2 * 169, em_b1 + 2 * 13, em_w2 + 2 * 78, em_b2 + 2 * 6,
        em_g + 2 * 6, em_bt + 2 * 6, (float*)d_out, n_pts);
}